// Readout_65755949302024
// MI455X (gfx1250) — compile-verified
//
#include <hip/hip_runtime.h>
#include <hip/hip_bf16.h>

// ---------------------------------------------------------------------------
// One-pass segmented softmax-readout for MI455X (gfx1250).
//   numer[g,f] = sum_{a in g} x * exp(x)      (accumulated in d_out)
//   denom[g,f] = sum_{a in g} exp(x)          (accumulated in d_ws)
//   out[g,f]   = numer / denom
// Streaming path: TDM (tensor_load_to_lds) double-buffered into LDS,
// synchronized with s_wait_tensorcnt; sorted indices -> register accumulation
// with rare atomic flushes.
// Roofline: 512 MB single pass @ 23.3 TB/s ~ 22 us (vs 44 us two-pass).
// ---------------------------------------------------------------------------

#define WAVES_PER_BLOCK 8
#define TILE_ROWS 16          // rows per TDM tile (16 * 256B = 4KB per DMA)
#define ROWS_PER_WAVE 256     // contiguous rows owned by one wave
#define FEAT 64

typedef unsigned int u32x4 __attribute__((ext_vector_type(4)));
typedef int          i32x8 __attribute__((ext_vector_type(8)));
typedef int          i32x4 __attribute__((ext_vector_type(4)));

typedef __attribute__((address_space(3))) const char lds_char_t;

static __device__ __forceinline__ unsigned lds_byte_offset(const void* p) {
    // generic -> LDS addrspace cast; AS(3) pointers are 32-bit LDS byte offsets
    return (unsigned)(unsigned long long)(lds_char_t*)p;
}

// Issue one TDM load: `rows` consecutive 64-float rows from global -> LDS.
// D# built per CDNA5 ISA ch.8 (group0: count/lds/global/type, group1: dims).
// tensor_dim == tile_dim so no OOB zero-fill (exp(0)=1 would corrupt denom).
static __device__ __forceinline__ void tdm_load_rows(unsigned lds_addr,
                                                     unsigned long long gaddr,
                                                     int rows) {
    unsigned la  = (unsigned)__builtin_amdgcn_readfirstlane((int)lds_addr);
    unsigned glo = (unsigned)__builtin_amdgcn_readfirstlane((int)(gaddr & 0xFFFFFFFFull));
    unsigned ghi = (unsigned)__builtin_amdgcn_readfirstlane((int)(gaddr >> 32));
    unsigned rw  = (unsigned)__builtin_amdgcn_readfirstlane(rows);

    u32x4 g0;
    g0[0] = 1u;                                  // count=1, user descriptor
    g0[1] = la;                                  // lds_addr (bytes)
    g0[2] = glo;                                 // global_addr[31:0]
    g0[3] = (ghi & 0x01FFFFFFu) | (2u << 30);    // global_addr[56:32] | type=2

    i32x8 g1;
    g1[0] = (int)(2u << 16);                     // workgroup_mask=0, data_size=2 (4B)
    g1[1] = (int)((unsigned)FEAT << 16);         // tensor_dim0 lo16 = 64
    g1[2] = (int)(rw << 16);                     // dim0 hi16=0 | tensor_dim1 lo16 = rows
    g1[3] = (int)((unsigned)FEAT << 16);         // dim1 hi16=0 | tile_dim0 = 64
    g1[4] = (int)rw;                             // tile_dim1 = rows | tile_dim2 = 0
    g1[5] = FEAT;                                // tensor_dim0_stride = 64 (contiguous rows)
    g1[6] = 0;                                   // stride0 hi | stride1 lo
    g1[7] = 0;                                   // stride1 hi

    i32x4 gz4 = {0, 0, 0, 0};                    // groups 2/3 unused (2-D tensor)
    i32x8 gz8 = {0, 0, 0, 0, 0, 0, 0, 0};        // 6-arg form extra group (zero)

    __builtin_amdgcn_tensor_load_to_lds(g0, g1, gz4, gz4, gz8, 0);
}

__global__ void __launch_bounds__(WAVES_PER_BLOCK * 32)
readout_accum(const float* __restrict__ feat, const int* __restrict__ gidx,
              float* __restrict__ numer, float* __restrict__ denom, int n_rows) {
    __shared__ float smem[WAVES_PER_BLOCK * 2 * TILE_ROWS * FEAT];  // 64 KB

    const int lane = threadIdx.x & 31;
    const int wave = threadIdx.x >> 5;
    const long long wid = (long long)blockIdx.x * WAVES_PER_BLOCK + wave;

    long long row_beg = wid * ROWS_PER_WAVE;
    if (row_beg >= n_rows) return;
    long long row_end = row_beg + ROWS_PER_WAVE;
    if (row_end > n_rows) row_end = n_rows;

    float* wbase = &smem[wave * (2 * TILE_ROWS * FEAT)];
    const unsigned lds0 = lds_byte_offset(wbase);
    const unsigned lds1 = lds0 + TILE_ROWS * FEAT * (unsigned)sizeof(float);

    const int n_tiles = (int)((row_end - row_beg + TILE_ROWS - 1) / TILE_ROWS);

    // Prologue: stage tile 0 into buffer 0.
    {
        long long rem = row_end - row_beg;
        int rows0 = rem < TILE_ROWS ? (int)rem : TILE_ROWS;
        tdm_load_rows(lds0,
                      (unsigned long long)(const void*)(feat + row_beg * FEAT),
                      rows0);
    }

    float aNx = 0.f, aNy = 0.f, aDx = 0.f, aDy = 0.f;
    int cur_g = -1;
    int buf = 0;

    for (int t = 0; t < n_tiles; ++t) {
        long long r0 = row_beg + (long long)t * TILE_ROWS;
        long long rem = row_end - r0;
        int rows = rem < TILE_ROWS ? (int)rem : TILE_ROWS;

        if (t + 1 < n_tiles) {
            long long r1 = r0 + TILE_ROWS;
            long long rem1 = row_end - r1;
            int rows1 = rem1 < TILE_ROWS ? (int)rem1 : TILE_ROWS;
            tdm_load_rows(buf ? lds0 : lds1,
                          (unsigned long long)(const void*)(feat + r1 * FEAT),
                          rows1);
            __builtin_amdgcn_s_wait_tensorcnt(1);  // oldest (current tile) done
        } else {
            __builtin_amdgcn_s_wait_tensorcnt(0);  // drain
        }

        // One coalesced index load per tile; broadcast per-row via shuffle.
        int gv = 0;
        if (lane < rows) gv = gidx[r0 + lane];

        const float* bptr = wbase + (buf ? TILE_ROWS * FEAT : 0);

        for (int r = 0; r < rows; ++r) {
            int g = __shfl(gv, r, 32);
            if (g != cur_g) {
                if (cur_g >= 0) {
                    float* o = numer + (size_t)cur_g * FEAT + lane * 2;
                    atomicAdd(o, aNx);
                    atomicAdd(o + 1, aNy);
                    float* d = denom + (size_t)cur_g * FEAT + lane * 2;
                    atomicAdd(d, aDx);
                    atomicAdd(d + 1, aDy);
                }
                cur_g = g;
                aNx = aNy = aDx = aDy = 0.f;
            }
            const float2 v = *(const float2*)(bptr + r * FEAT + lane * 2);
            float ex = __expf(v.x);
            float ey = __expf(v.y);
            aDx += ex;
            aDy += ey;
            aNx = fmaf(v.x, ex, aNx);
            aNy = fmaf(v.y, ey, aNy);
        }
        buf ^= 1;
    }

    if (cur_g >= 0) {
        float* o = numer + (size_t)cur_g * FEAT + lane * 2;
        atomicAdd(o, aNx);
        atomicAdd(o + 1, aNy);
        float* d = denom + (size_t)cur_g * FEAT + lane * 2;
        atomicAdd(d, aDx);
        atomicAdd(d + 1, aDy);
    }
}

__global__ void zero2_kernel(float* __restrict__ a, float* __restrict__ b, int n) {
    int i = blockIdx.x * blockDim.x + threadIdx.x;
    if (i < n) {
        a[i] = 0.f;
        b[i] = 0.f;
    }
}

__global__ void finalize_kernel(float* __restrict__ out,
                                const float* __restrict__ den, int n) {
    int i = blockIdx.x * blockDim.x + threadIdx.x;
    if (i < n) {
        float d = den[i];
        out[i] = (d != 0.f) ? (out[i] / d) : 0.f;  // empty segment -> 0 (matches segment_sum)
    }
}

extern "C" void kernel_launch(void* const* d_in, const int* in_sizes, int n_in,
                              void* d_out, int out_size, void* d_ws, size_t ws_size,
                              hipStream_t stream) {
    (void)n_in;
    (void)ws_size;

    const float* feat = (const float*)d_in[0];   // (N_atoms, 64) f32
    const int* gidx   = (const int*)d_in[1];     // (N_atoms,) i32, sorted
    const int n_rows  = in_sizes[1];
    const int gtot    = out_size;                // num_graphs * 64

    float* numer = (float*)d_out;                // accumulate numerator in-place
    float* denom = (float*)d_ws;                 // 4 MB scratch for denominators

    {
        int thr = 256;
        int blk = (gtot + thr - 1) / thr;
        zero2_kernel<<<blk, thr, 0, stream>>>(numer, denom, gtot);
    }
    {
        long long waves  = ((long long)n_rows + ROWS_PER_WAVE - 1) / ROWS_PER_WAVE;
        int       blocks = (int)((waves + WAVES_PER_BLOCK - 1) / WAVES_PER_BLOCK);
        readout_accum<<<blocks, WAVES_PER_BLOCK * 32, 0, stream>>>(
            feat, gidx, numer, denom, n_rows);
    }
    {
        int thr = 256;
        int blk = (gtot + thr - 1) / thr;
        finalize_kernel<<<blk, thr, 0, stream>>>(numer, denom, gtot);
    }
}